// PixelFlow_66271345377404
// MI455X (gfx1250) — compile-verified
//
#include <hip/hip_runtime.h>
#include <stdint.h>

// PixelFlow: bilinear backward-warp + visibility gating.
// img:   [B,512,512,3] f32, flows: [B,512,512,3] f32 -> out [B,512,512,3] f32
// Memory-bound (~150MB traffic, ~6.5us floor at 23.3 TB/s). Whole working set
// fits in the 192MB L2. gfx1250 path: async global->LDS b128 staging of the
// 12B/pixel flow stream (ASYNCcnt + s_wait_asynccnt), b96 texel gathers,
// LDS repack + coalesced b128 stores.

#define HW 512
#define HWSHIFT 9
#define PIX_PER_BLK 256
#define CHUNKS 192            // 256 pixels * 3 floats = 768 floats = 192 float4

struct __align__(4) F3 { float x, y, z; };

__device__ __forceinline__ void warp_one(const float* __restrict__ img,
                                         int p, float fx, float fy, float vis,
                                         float& r0, float& r1, float& r2) {
  const int rem = p & (HW * HW - 1);
  const int b   = p >> (2 * HWSHIFT);
  const int yi  = rem >> HWSHIFT;
  const int xi  = rem & (HW - 1);

  // match reference: normalize to [-1,1] then unnormalize to [0, W-1]
  const float gx = 2.0f * (((float)xi + fx) * (1.0f / (float)HW) - 0.5f);
  const float gy = 2.0f * (((float)yi + fy) * (1.0f / (float)HW) - 0.5f);
  const float px = 0.5f * (gx + 1.0f) * (float)(HW - 1);
  const float py = 0.5f * (gy + 1.0f) * (float)(HW - 1);

  int x0 = (int)floorf(px);
  int y0 = (int)floorf(py);
  int x1 = x0 + 1;
  int y1 = y0 + 1;
  x0 = min(max(x0, 0), HW - 1);
  x1 = min(max(x1, 0), HW - 1);
  y0 = min(max(y0, 0), HW - 1);
  y1 = min(max(y1, 0), HW - 1);

  // weights from the *clamped* integer coords (reference semantics)
  const float x0f = (float)x0, x1f = (float)x1;
  const float y0f = (float)y0, y1f = (float)y1;
  const float wa = (x1f - px) * (y1f - py);
  const float wb = (x1f - px) * (py - y0f);
  const float wc = (px - x0f) * (y1f - py);
  const float wd = (px - x0f) * (py - y0f);

  const F3* __restrict__ plane =
      (const F3*)(img + (size_t)b * (size_t)(HW * HW * 3));
  const F3 Ia = plane[(size_t)(y0 * HW + x0)];   // b96 gathers, L2-resident
  const F3 Ib = plane[(size_t)(y1 * HW + x0)];
  const F3 Ic = plane[(size_t)(y0 * HW + x1)];
  const F3 Id = plane[(size_t)(y1 * HW + x1)];

  const float s = tanhf(vis) + 1.0f;
  r0 = (wa * Ia.x + wb * Ib.x + wc * Ic.x + wd * Id.x) * s;
  r1 = (wa * Ia.y + wb * Ib.y + wc * Ic.y + wd * Id.y) * s;
  r2 = (wa * Ia.z + wb * Ib.z + wc * Ic.z + wd * Id.z) * s;
}

__global__ __launch_bounds__(PIX_PER_BLK)
void pixelflow_kernel(const float* __restrict__ img,
                      const float* __restrict__ flows,
                      float* __restrict__ out, int n_pix) {
  __shared__ __align__(16) float smem[PIX_PER_BLK * 3];

  const int t  = threadIdx.x;
  const int p0 = blockIdx.x * PIX_PER_BLK;
  const int p  = p0 + t;
  const bool full = (p0 + PIX_PER_BLK) <= n_pix;   // uniform per block

  if (full) {
    // ---- stage 1: async-copy this block's 3KB flow window into LDS ----
    if (t < CHUNKS) {
      const unsigned lds = (unsigned)(uintptr_t)(&smem[t * 4]); // LDS byte off
      const float* gsrc = flows + (size_t)p0 * 3 + (size_t)t * 4; // 16B aligned
      asm volatile("global_load_async_to_lds_b128 %0, %1, off"
                   :: "v"(lds), "v"(gsrc) : "memory");
    }
    asm volatile("s_wait_asynccnt 0" ::: "memory");
    __syncthreads();

    const float fx  = smem[t * 3 + 0];   // stride-3: conflict-free (gcd(3,64)=1)
    const float fy  = smem[t * 3 + 1];
    const float vis = smem[t * 3 + 2];

    float r0, r1, r2;
    warp_one(img, p, fx, fy, vis, r0, r1, r2);

    // ---- stage 2: repack results in LDS, emit coalesced b128 stores ----
    __syncthreads();               // all flow reads done before overwrite
    smem[t * 3 + 0] = r0;
    smem[t * 3 + 1] = r1;
    smem[t * 3 + 2] = r2;
    __syncthreads();
    if (t < CHUNKS) {
      const float4 v = *(const float4*)&smem[t * 4];
      *(float4*)(out + (size_t)p0 * 3 + (size_t)t * 4) = v;
    }
  } else {
    // tail block: scalar path, no barriers needed
    if (p < n_pix) {
      const float fx  = flows[(size_t)p * 3 + 0];
      const float fy  = flows[(size_t)p * 3 + 1];
      const float vis = flows[(size_t)p * 3 + 2];
      float r0, r1, r2;
      warp_one(img, p, fx, fy, vis, r0, r1, r2);
      out[(size_t)p * 3 + 0] = r0;
      out[(size_t)p * 3 + 1] = r1;
      out[(size_t)p * 3 + 2] = r2;
    }
  }
}

extern "C" void kernel_launch(void* const* d_in, const int* in_sizes, int n_in,
                              void* d_out, int out_size, void* d_ws, size_t ws_size,
                              hipStream_t stream) {
  (void)n_in; (void)out_size; (void)d_ws; (void)ws_size;
  const float* img   = (const float*)d_in[0];
  const float* flows = (const float*)d_in[1];
  float* out = (float*)d_out;

  const int n_pix  = in_sizes[1] / 3;                 // B*H*W
  const int blocks = (n_pix + PIX_PER_BLK - 1) / PIX_PER_BLK;
  pixelflow_kernel<<<blocks, PIX_PER_BLK, 0, stream>>>(img, flows, out, n_pix);
}